// CombinedLoss_88450556494045
// MI455X (gfx1250) — compile-verified
//
#include <hip/hip_runtime.h>

// CDNA5 (gfx1250) implementation of the 5-level anchor->annotation assignment.
// Memory-bound (12.4 MB out, ~1 MB in); per-block the 3KB annotation table is
// staged to LDS via the CDNA5 async global->LDS data mover, derived values are
// precomputed into SoA LDS arrays, and each thread does a broadcast-read
// predicated argmin over annotations for its anchor.

#if defined(__has_builtin)
#  if __has_builtin(__builtin_amdgcn_global_load_async_to_lds_b128) && \
      __has_builtin(__builtin_amdgcn_s_wait_asynccnt)
#    define USE_ASYNC_LDS 1
#  endif
#endif

typedef int v4i __attribute__((ext_vector_type(4)));
typedef __attribute__((address_space(1))) v4i gas_v4i;  // global int4
typedef __attribute__((address_space(3))) v4i las_v4i;  // LDS int4

__global__ __launch_bounds__(256) void assign_level_kernel(
    const float* __restrict__ ann,      // 256 x 3 (l, r, cls), l sorted ascending
    const float* __restrict__ anchors,  // n anchors for this level
    float* __restrict__ out,            // full concatenated output buffer
    int n, int anchor_off, int at_total,
    float stride, float lo, float hi, float level)
{
    __shared__ __align__(16) float raw[768];   // staged annotation table (AoS)
    __shared__ float sl[256];                  // l
    __shared__ float sr[256];                  // r
    __shared__ float srmin[256];               // min(r, l + radius*stride)
    __shared__ float sarea[256];               // r - l

    const int tid = threadIdx.x;

    // ---- Stage 3KB annotation table into LDS (async global->LDS data mover) ----
#if defined(USE_ASYNC_LDS)
    if (tid < 192) {  // 192 lanes x 16B = 3072B
        const v4i* gsrc = (const v4i*)ann + tid;  // generic pointers...
        v4i*       ldst = (v4i*)raw + tid;
        __builtin_amdgcn_global_load_async_to_lds_b128(
            (gas_v4i*)gsrc, (las_v4i*)ldst, /*offset=*/0, /*cpol=*/0);
    }
    __builtin_amdgcn_s_wait_asynccnt(0);
#else
    for (int t = tid; t < 768; t += 256) raw[t] = ann[t];
#endif
    __syncthreads();

    // ---- Precompute per-annotation derived values (one annotation per thread) ----
    {
        float lj = raw[3 * tid + 0];
        float rj = raw[3 * tid + 1];
        float cj = raw[3 * tid + 2];
        // radius = (cls==0)*4.5 + (cls==1)*2.5   (cls is exactly 0.0 or 1.0)
        float radius = (cj == 0.0f ? 4.5f : 0.0f) + (cj == 1.0f ? 2.5f : 0.0f);
        float limit  = lj + radius * stride;
        sl[tid]    = lj;
        sr[tid]    = rj;
        srmin[tid] = fminf(rj, limit);
        sarea[tid] = rj - lj;
    }
    __syncthreads();

    const int i = blockIdx.x * 256 + tid;
    if (i >= n) return;

    const float a = anchors[i];
    const float INF_F = 100000000.0f;  // exactly representable in f32

    float best = INF_F;
    int   bidx = 0;
    for (int j = 0; j < 256; ++j) {
        float lj = sl[j];
        if (lj > a) break;  // l is sorted: no later annotation can contain a
        float rm  = srmin[j];
        float rj  = sr[j];
        float lsj = a - lj;
        float rsj = rj - a;
        float m   = fmaxf(lsj, rsj);
        bool cond = (a >= lj) && (a <= rm) && (m >= lo) && (m <= hi);
        float cand = cond ? sarea[j] : INF_F;
        if (cand < best) { best = cand; bidx = j; }  // strict < => first-min (argmin)
    }
    const bool pos = (best != INF_F);

    // Chosen (or default index 0) annotation
    const float al  = sl[bidx];
    const float arr = sr[bidx];
    const float ac  = pos ? raw[3 * bidx + 2] : 0.0f;

    const float lsv = a - al;
    const float rsv = arr - a;
    const float inv_s = 1.0f / stride;  // stride is 2^k: reciprocal exact, mul == div bitwise

    const size_t AT = (size_t)at_total;
    const size_t g  = (size_t)anchor_off + (size_t)i;

    float* out_pos = out;
    float* out_asg = out + AT;
    float* out_nrm = out + 4 * AT;
    float* out_ls  = out + 7 * AT;
    float* out_rs  = out + 8 * AT;
    float* out_nls = out + 9 * AT;
    float* out_nrs = out + 10 * AT;
    float* out_lvl = out + 11 * AT;

    out_pos[g] = pos ? 1.0f : 0.0f;

    out_asg[3 * g + 0] = al;
    out_asg[3 * g + 1] = arr;
    out_asg[3 * g + 2] = ac;

    out_nrm[3 * g + 0] = al * inv_s;
    out_nrm[3 * g + 1] = arr * inv_s;
    out_nrm[3 * g + 2] = ac;

    out_ls[g]  = lsv;
    out_rs[g]  = rsv;
    out_nls[g] = lsv * inv_s;
    out_nrs[g] = rsv * inv_s;
    out_lvl[g] = level;
}

extern "C" void kernel_launch(void* const* d_in, const int* in_sizes, int n_in,
                              void* d_out, int out_size, void* d_ws, size_t ws_size,
                              hipStream_t stream) {
    (void)d_ws; (void)ws_size; (void)out_size; (void)n_in;

    const float* ann = (const float*)d_in[0];
    float* out = (float*)d_out;

    // Total anchors across the 5 levels (output section stride).
    int at_total = 0;
    for (int lvl = 0; lvl < 5; ++lvl) at_total += in_sizes[1 + lvl];

    // Replicate the reference's float64 constant construction exactly, then
    // cast to f32 (JAX compares f32 arrays against weak Python-float scalars).
    const double TR = 22050.0 / 256.0;  // 86.1328125 exact
    const double e0 = (0.35 + 0.7) / 2.0;
    const double e1 = (0.7 + 1.4) / 2.0;
    const double e2 = (1.4 + 2.8) / 2.0;
    const double e3 = (2.8 + 5.6) / 2.0;
    const double loD[5] = { -1.0, e0, e1, e2, e3 };
    const double hiD[5] = { e0, e1, e2, e3, 1000.0 };

    int off = 0;
    for (int lvl = 0; lvl < 5; ++lvl) {
        const int n = in_sizes[1 + lvl];
        const float stride = (float)(1 << lvl);
        const float lo = (float)(loD[lvl] * TR);
        const float hi = (float)(hiD[lvl] * TR);
        const float level = (float)(lvl + 1);

        dim3 grid((unsigned)((n + 255) / 256));
        assign_level_kernel<<<grid, 256, 0, stream>>>(
            ann, (const float*)d_in[1 + lvl], out,
            n, off, at_total, stride, lo, hi, level);
        off += n;
    }
}